// EdgeNetwork_11776800325765
// MI455X (gfx1250) — compile-verified
//
#include <hip/hip_runtime.h>

typedef __attribute__((ext_vector_type(2))) float v2f;
typedef __attribute__((ext_vector_type(8))) float v8f;

struct alignas(8) i2 { int x, y; };
struct alignas(8) f2 { float x, y; };

// One wave processes 16 edges per iteration with 4 chained
// V_WMMA_F32_16X16X4_F32 ops:
//   D(16x16) = A(16x4 per chunk) x B(4x16 per chunk), accumulated over 4 chunks
//   A[i][c] = W[c][i], W[k*4+j][i] = K[k][i*4+j]  (chunk 3: bias row, "ef=1")
//   B[c][n] = O[n][c], O[n][k*4+j] = ef[n][k]*src[n][j] (chunk 3: src[n][j])
//   => D[i][n] = transformed[edge n][i], lane-local in lanes 0..15.
__global__ __launch_bounds__(256)
void edgenet_wmma_kernel(const float* __restrict__ nf,
                         const float* __restrict__ ef,
                         const int*   __restrict__ pair,
                         const float* __restrict__ kern,
                         const float* __restrict__ bias,
                         float* __restrict__ out,
                         int E, int ntiles)
{
    const int  lane = threadIdx.x & 31;   // wave32
    const bool hi   = lane >= 16;
    const int  i    = lane & 15;
    const int  joff = hi ? 2 : 0;

    // Loop-invariant A operand (W^T). Rows i>=4 are zero.
    // A layout (16x4 f32): lanes 0-15: V0=K0,V1=K1 ; lanes 16-31: V0=K2,V1=K3.
    v2f a0 = {0.f, 0.f}, a1 = a0, a2 = a0, a3 = a0;
    if (i < 4) {
        const int col = i * 4 + joff;
        a0 = v2f{ kern[ 0 + col], kern[ 0 + col + 1] };
        a1 = v2f{ kern[16 + col], kern[16 + col + 1] };
        a2 = v2f{ kern[32 + col], kern[32 + col + 1] };
        a3 = v2f{ bias[col],      bias[col + 1] };
    }

    const int wave = (blockIdx.x * blockDim.x + threadIdx.x) >> 5;
    const int nwav = (gridDim.x * blockDim.x) >> 5;

    for (int t = wave; t < ntiles; t += nwav) {
        const int  e     = t * 16 + i;          // both lane halves: same edge
        const bool valid = e < E;

        i2 idx = {0, 0};
        float e0 = 0.f, e1 = 0.f, e2 = 0.f;
        f2 s = {0.f, 0.f};
        if (valid) {
            idx = *(const i2*)(pair + 2 * (long long)e);
            const float* ep = ef + 3 * (long long)e;
            e0 = ep[0]; e1 = ep[1]; e2 = ep[2];
            // lower lanes need src[0..1], upper lanes src[2..3]; 8B aligned
            s = *(const f2*)(nf + 4 * (long long)idx.x + joff);
        }

        // Prefetch next iteration's edge stream (global_prefetch_b8).
        const int tn = t + nwav;
        if (tn < ntiles) {
            const long long en = (long long)tn * 16 + i;
            __builtin_prefetch(ef + 3 * en, 0, 0);
            __builtin_prefetch(pair + 2 * en, 0, 0);
        }

        // B layout (4x16 f32): lanes 0-15: V0=K0,V1=K1 ; lanes 16-31: V0=K2,V1=K3.
        v2f b0 = v2f{ e0 * s.x, e0 * s.y };   // O[n][0..3]   (k=0)
        v2f b1 = v2f{ e1 * s.x, e1 * s.y };   // O[n][4..7]   (k=1)
        v2f b2 = v2f{ e2 * s.x, e2 * s.y };   // O[n][8..11]  (k=2)
        v2f b3 = v2f{      s.x,      s.y };   // O[n][12..15] (bias row)

        v8f d = {0.f, 0.f, 0.f, 0.f, 0.f, 0.f, 0.f, 0.f};
        d = __builtin_amdgcn_wmma_f32_16x16x4_f32(false, a0, false, b0, (short)0, d, false, false);
        d = __builtin_amdgcn_wmma_f32_16x16x4_f32(false, a1, false, b1, (short)0, d, false, false);
        d = __builtin_amdgcn_wmma_f32_16x16x4_f32(false, a2, false, b2, (short)0, d, false, false);
        d = __builtin_amdgcn_wmma_f32_16x16x4_f32(false, a3, false, b3, (short)0, d, false, false);

        // D VGPR v, lane L<16 == transformed[edge L][v]; scatter is lane-local.
        if (!hi && valid) {
            float* op = out + 4 * (long long)idx.y;
            unsafeAtomicAdd(op + 0, d[0]);
            unsafeAtomicAdd(op + 1, d[1]);
            unsafeAtomicAdd(op + 2, d[2]);
            unsafeAtomicAdd(op + 3, d[3]);
        }
    }
}

extern "C" void kernel_launch(void* const* d_in, const int* in_sizes, int n_in,
                              void* d_out, int out_size, void* d_ws, size_t ws_size,
                              hipStream_t stream) {
    const float* nf   = (const float*)d_in[0];   // node_features  (N,4) f32
    const float* ef   = (const float*)d_in[1];   // edge_features  (E,3) f32
    const int*   pair = (const int*)  d_in[2];   // pair_indices   (E,2) int
    const float* kern = (const float*)d_in[3];   // kernel         (3,16) f32
    const float* bias = (const float*)d_in[4];   // bias           (16,) f32
    float* out = (float*)d_out;

    const int E      = in_sizes[1] / 3;
    const int ntiles = (E + 15) / 16;

    // Output is accumulated with atomics; harness poisons d_out -> zero it.
    hipMemsetAsync(d_out, 0, (size_t)out_size * sizeof(float), stream);

    const int threads = 256;                 // 8 waves / block
    int blocks = 2048;                       // 16384 waves, ~61 tiles each
    const long long maxb = ((long long)ntiles * 32 + threads - 1) / threads;
    if (blocks > maxb) blocks = (int)maxb;
    if (blocks < 1) blocks = 1;

    edgenet_wmma_kernel<<<blocks, threads, 0, stream>>>(nf, ef, pair, kern, bias,
                                                        out, E, ntiles);
}